// RepeatEncoder_53480932769965
// MI455X (gfx1250) — compile-verified
//
#include <hip/hip_runtime.h>

// ---------------------------------------------------------------------------
// RepeatEncoder / BN / LIF spiking network, MI455X (gfx1250, wave32).
//
// Math collapses: input is broadcast over the 16 BN channels, so per-channel
// batch stats == global mean/var of the whole input. Then a per-element
// 16-step LIF scan. Memory-bound: ~16 MB read + 128 MB written -> ~6 us at
// 23.3 TB/s. Reduction uses V_WMMA_F32_16X16X4_F32 with an all-ones A matrix
// (row0 of D = column sums of B; total independent of B lane layout),
// keeping full f32 precision on the matrix pipe.
// ---------------------------------------------------------------------------

typedef float v2f __attribute__((ext_vector_type(2)));
typedef float v4f __attribute__((ext_vector_type(4)));
typedef float v8f __attribute__((ext_vector_type(8)));

#define N_ELEM     (64 * 128 * 256)   // 2,097,152 input elements
#define NV4        (N_ELEM / 4)       // 524,288 float4 elements
#define S_STEPS    16
#define THREADS    256
#define RED_BLOCKS 512                // 512*256 threads * 4 iters * 4 elems = N_ELEM
#define RED_ITERS  (NV4 / (RED_BLOCKS * THREADS))   // = 4, no tail
#define EPS        1e-5f

// ---------------------------------------------------------------------------
// Pass 1: global sum / sum-of-squares via WMMA f32 16x16x4 with A = ones.
// D[m,n] = sum_k B[k,n] + C[m,n]; all rows identical, so summing D's VGPR0
// across all 32 lanes (row0 cols + row8 cols) yields 2 * total.
// Deterministic: fixed iteration order, fixed shuffle tree, fixed LDS tree.
// ---------------------------------------------------------------------------
__global__ __launch_bounds__(THREADS) void snn_reduce_kernel(
    const v4f* __restrict__ in4, float* __restrict__ partials)
{
    v8f acc  = {};   // sum accumulator (16x16 f32 D/C)
    v8f accq = {};   // sum-of-squares accumulator
    const v2f ones = {1.0f, 1.0f};

    const int tid    = blockIdx.x * THREADS + threadIdx.x;
    const int stride = RED_BLOCKS * THREADS;

    #pragma unroll
    for (int it = 0; it < RED_ITERS; ++it) {
        v4f x = in4[tid + it * stride];
        v2f lo  = {x.x, x.y};
        v2f hi  = {x.z, x.w};
        v2f loq = lo * lo;
        v2f hiq = hi * hi;
        acc  = __builtin_amdgcn_wmma_f32_16x16x4_f32(false, ones, false, lo,
                                                     (short)0, acc,  false, false);
        acc  = __builtin_amdgcn_wmma_f32_16x16x4_f32(false, ones, false, hi,
                                                     (short)0, acc,  false, false);
        accq = __builtin_amdgcn_wmma_f32_16x16x4_f32(false, ones, false, loq,
                                                     (short)0, accq, false, false);
        accq = __builtin_amdgcn_wmma_f32_16x16x4_f32(false, ones, false, hiq,
                                                     (short)0, accq, false, false);
    }

    // VGPR0 of D: lanes 0-15 = row0 col sums, lanes 16-31 = row8 col sums.
    float s = acc[0];
    float q = accq[0];
    #pragma unroll
    for (int m = 16; m >= 1; m >>= 1) {
        s += __shfl_xor(s, m, 32);
        q += __shfl_xor(q, m, 32);
    }
    s *= 0.5f;   // lane-sum covered rows 0 and 8 -> 2x total
    q *= 0.5f;

    __shared__ float ls[THREADS / 32];
    __shared__ float lq[THREADS / 32];
    const int wave = threadIdx.x >> 5;
    const int lane = threadIdx.x & 31;
    if (lane == 0) { ls[wave] = s; lq[wave] = q; }
    __syncthreads();
    if (threadIdx.x == 0) {
        float S = 0.0f, Q = 0.0f;
        #pragma unroll
        for (int w = 0; w < THREADS / 32; ++w) { S += ls[w]; Q += lq[w]; }
        partials[blockIdx.x]              = S;
        partials[RED_BLOCKS + blockIdx.x] = Q;
    }
}

// ---------------------------------------------------------------------------
// Pass 2: fold 512 per-block partials (fixed tree -> deterministic),
// emit mean and rstd = rsqrt(var + eps).
// ---------------------------------------------------------------------------
__global__ __launch_bounds__(THREADS) void snn_finalize_kernel(
    const float* __restrict__ partials, float* __restrict__ stats)
{
    __shared__ float ss[THREADS];
    __shared__ float sq[THREADS];
    const int t = threadIdx.x;
    ss[t] = partials[t] + partials[t + THREADS];
    sq[t] = partials[RED_BLOCKS + t] + partials[RED_BLOCKS + t + THREADS];
    __syncthreads();
    #pragma unroll
    for (int off = THREADS / 2; off >= 1; off >>= 1) {
        if (t < off) { ss[t] += ss[t + off]; sq[t] += sq[t + off]; }
        __syncthreads();
    }
    if (t == 0) {
        const float inv_n = 1.0f / (float)N_ELEM;
        const float mean  = ss[0] * inv_n;
        const float var   = sq[0] * inv_n - mean * mean;
        stats[0] = mean;
        stats[1] = rsqrtf(var + EPS);
    }
}

// ---------------------------------------------------------------------------
// Pass 3: normalize + 16-step LIF scan held in registers.
// Per thread: 1 coalesced b128 load, 16 coalesced non-temporal b128 stores.
// out[s*N + i] : spikes[s, b, d1, d2] with [b,d1,d2] flat == input layout.
// ---------------------------------------------------------------------------
__global__ __launch_bounds__(THREADS) void snn_main_kernel(
    const v4f*   __restrict__ in4,
    const float* __restrict__ gamma,
    const float* __restrict__ beta,
    const float* __restrict__ thrp,
    const float* __restrict__ stats,
    v4f*         __restrict__ out4)
{
    const int   i    = blockIdx.x * THREADS + threadIdx.x;
    const float mean = stats[0];
    const float rstd = stats[1];
    const float thr  = thrp[0];

    v4f x = in4[i];
    const float c0 = (x.x - mean) * rstd;
    const float c1 = (x.y - mean) * rstd;
    const float c2 = (x.z - mean) * rstd;
    const float c3 = (x.w - mean) * rstd;

    float v0 = 0.0f, v1 = 0.0f, v2 = 0.0f, v3 = 0.0f;

    #pragma unroll
    for (int s = 0; s < S_STEPS; ++s) {
        const float g = gamma[s];   // uniform address -> scalar loads, L2-hot
        const float b = beta[s];
        // x_s = gamma*(x-mean)*rstd + beta
        const float x0 = g * c0 + b;
        const float x1 = g * c1 + b;
        const float x2 = g * c2 + b;
        const float x3 = g * c3 + b;
        // neuronal charge: v += (x - v)/tau, tau = 2 (mirror reference order)
        v0 = v0 + (x0 - v0) * 0.5f;
        v1 = v1 + (x1 - v1) * 0.5f;
        v2 = v2 + (x2 - v2) * 0.5f;
        v3 = v3 + (x3 - v3) * 0.5f;
        // fire + hard reset to 0
        const float s0 = (v0 >= thr) ? 1.0f : 0.0f;
        const float s1 = (v1 >= thr) ? 1.0f : 0.0f;
        const float s2 = (v2 >= thr) ? 1.0f : 0.0f;
        const float s3 = (v3 >= thr) ? 1.0f : 0.0f;
        v0 = (v0 >= thr) ? 0.0f : v0;
        v1 = (v1 >= thr) ? 0.0f : v1;
        v2 = (v2 >= thr) ? 0.0f : v2;
        v3 = (v3 >= thr) ? 0.0f : v3;

        v4f o = {s0, s1, s2, s3};
        // 128 MB write-once stream: non-temporal to keep L2 clean
        __builtin_nontemporal_store(o, out4 + (size_t)s * NV4 + i);
    }
}

// ---------------------------------------------------------------------------
extern "C" void kernel_launch(void* const* d_in, const int* in_sizes, int n_in,
                              void* d_out, int out_size, void* d_ws, size_t ws_size,
                              hipStream_t stream)
{
    (void)in_sizes; (void)n_in; (void)out_size; (void)ws_size;

    const v4f*   in4   = (const v4f*)d_in[0];   // [64,128,256] f32
    const float* gamma = (const float*)d_in[1]; // [16]
    const float* beta  = (const float*)d_in[2]; // [16]
    const float* thr   = (const float*)d_in[3]; // [1]

    float* partials = (float*)d_ws;                 // 2*512 floats
    float* stats    = partials + 2 * RED_BLOCKS;    // mean, rstd

    snn_reduce_kernel  <<<RED_BLOCKS, THREADS, 0, stream>>>(in4, partials);
    snn_finalize_kernel<<<1,          THREADS, 0, stream>>>(partials, stats);
    snn_main_kernel    <<<NV4 / THREADS, THREADS, 0, stream>>>(
        in4, gamma, beta, thr, stats, (v4f*)d_out);
}